// Indexed_LinearFC_List_32667521254078
// MI455X (gfx1250) — compile-verified
//
#include <hip/hip_runtime.h>

// ---- problem constants (match reference) ----
#define NUM_MODULES 64
#define INP 256
#define OUTF 256
#define BATCH 2048
#define TSEQ 64
#define XP 258          // padded LDS row stride (floats): 256 data + 2 pad (TDM pad fields)
#define CH_ROWS 128     // W rows per chunk
#define WBUF_DW (CH_ROWS * XP)          // 33024 dwords per W buffer
#define LDS_BYTES (2 * WBUF_DW * 4)     // 264192 B <= 320KB WGP budget

typedef __attribute__((ext_vector_type(2))) float        v2f;
typedef __attribute__((ext_vector_type(8))) float        v8f;
typedef __attribute__((ext_vector_type(4))) unsigned int v4u;
typedef __attribute__((ext_vector_type(8))) int          v8i;
typedef __attribute__((ext_vector_type(4))) int          v4i;

// Issue one TDM 2D tile load: rows x 256 f32, row-major global (stride 256),
// into LDS at lds_off with 2-dword padding after every 256 dwords (row stride XP=258).
// D# layout per CDNA5 ISA §8.3/8.4. 6-arg builtin form (clang-23 / therock-10.0).
__device__ __forceinline__ void tdm_load_2d(unsigned lds_off, const float* gsrc,
                                            unsigned rows) {
  size_t ga = (size_t)gsrc;
  v4u g0;
  g0[0] = 1u;                                       // count=1 (valid), no gather
  g0[1] = lds_off;                                  // lds_addr (bytes)
  g0[2] = (unsigned)(ga & 0xFFFFFFFFu);             // global_addr[31:0]
  g0[3] = (unsigned)((ga >> 32) & 0x01FFFFFFu)      // global_addr[56:32]
        | (2u << 30);                               // type=2 ("image")
  v8i g1;
  g1[0] = (int)((2u << 16)                          // data_size = 4B
              | (1u << 20)                          // pad_enable
              | (7u << 22)                          // pad_interval: 256 dwords
              | (1u << 25));                        // pad_amount: 2 dwords
  g1[1] = (int)(256u << 16);                        // tensor_dim0 = 256 (lo16); abar=0
  g1[2] = (int)(rows << 16);                        // tensor_dim1 lo16; tensor_dim0 hi=0
  g1[3] = (int)(256u << 16);                        // tile_dim0 = 256; tensor_dim1 hi=0
  g1[4] = (int)rows;                                // tile_dim1 = rows; tile_dim2=0
  g1[5] = 256;                                      // tensor_dim0_stride lo32
  g1[6] = 0;
  g1[7] = 0;
  v4i z4 = {0, 0, 0, 0};                            // groups 2/3 unused (2D tensor)
  v8i z8 = {0, 0, 0, 0, 0, 0, 0, 0};                // extra group storage (unused)
  __builtin_amdgcn_tensor_load_to_lds(g0, g1, z4, z4, z8, 0);
}

__device__ __forceinline__ void store_tile(float* __restrict__ yp, float bb, v8f a) {
#pragma unroll
  for (int v = 0; v < 8; ++v) yp[v * OUTF] = a[v] + bb;
}

__global__ __launch_bounds__(256)
void moe_fc_wmma_tdm(const float* __restrict__ x,
                     const int*   __restrict__ indices,
                     const float* __restrict__ W,
                     const float* __restrict__ bias,
                     float*       __restrict__ y) {
  extern __shared__ float smem[];
  float* Wb0 = smem;             // buffer 0: first X staging area, then W chunk 0
  float* Wb1 = smem + WBUF_DW;   // buffer 1: W chunk 1

  const int b   = blockIdx.x;
  const int tid = threadIdx.x;
  const int e   = indices[b];

  const float* xb = x + (size_t)b * TSEQ * INP;
  const float* We = W + (size_t)e * OUTF * INP;
  float*       yb = y + (size_t)b * TSEQ * OUTF;

  const int lane = tid & 31;
  const int wave = tid >> 5;
  const int rt   = wave & 3;     // row tile: t0 = rt*16
  const int cg   = wave >> 2;    // 0..1 -> 4 col tiles each
  const int hlf  = lane >> 4;    // K-pair select (0: K=0,1  1: K=2,3)
  const int l16  = lane & 15;

  const unsigned ldsBase = (unsigned)(size_t)(void*)smem;  // LDS aperture: low 32b = offset

  // ---- TDM: stage X[b] (64x256 f32, padded rows) into buffer 0 ----
  if (wave == 0) {
    tdm_load_2d(ldsBase, xb, TSEQ);
    __builtin_amdgcn_s_wait_tensorcnt(0);
  }
  __syncthreads();

  // ---- pull this wave's A-fragments into registers (ISA 16x4 f32 layout) ----
  const float* Arow = Wb0 + (rt * 16 + l16) * XP + 2 * hlf;
  v2f Areg[64];
#pragma unroll
  for (int s = 0; s < 64; ++s) Areg[s] = *(const v2f*)(Arow + 4 * s);
  __syncthreads();   // everyone done reading X before W chunk 0 overwrites it

  // ---- TDM: double-buffered W chunks (in-order per wave; wait for chunk 0 only) ----
  if (wave == 0) {
    tdm_load_2d(ldsBase,                      We,                 CH_ROWS);
    tdm_load_2d(ldsBase + WBUF_DW * 4,        We + CH_ROWS * INP, CH_ROWS);
    __builtin_amdgcn_s_wait_tensorcnt(1);     // chunk 0 landed; chunk 1 still in flight
  }
  __syncthreads();

  for (int oc = 0; oc < 2; ++oc) {
    const float* Wc = (oc == 0) ? Wb0 : Wb1;
    // B fragment bases: B[k][n] = W[o0+n][k]; n = l16, k offset 2*hlf
    const float* B0 = Wc + ((cg * 4 + 0) * 16 + l16) * XP + 2 * hlf;
    const float* B1 = B0 + 16 * XP;
    const float* B2 = B1 + 16 * XP;
    const float* B3 = B2 + 16 * XP;

    v8f acc0 = {}, acc1 = {}, acc2 = {}, acc3 = {};

#pragma unroll
    for (int s = 0; s < 64; ++s) {
      const int k = 4 * s;
      v2f a   = Areg[s];
      v2f bf0 = *(const v2f*)(B0 + k);
      v2f bf1 = *(const v2f*)(B1 + k);
      v2f bf2 = *(const v2f*)(B2 + k);
      v2f bf3 = *(const v2f*)(B3 + k);
      acc0 = __builtin_amdgcn_wmma_f32_16x16x4_f32(false, a, false, bf0,
                                                   (short)0, acc0, false, false);
      acc1 = __builtin_amdgcn_wmma_f32_16x16x4_f32(false, a, false, bf1,
                                                   (short)0, acc1, false, false);
      acc2 = __builtin_amdgcn_wmma_f32_16x16x4_f32(false, a, false, bf2,
                                                   (short)0, acc2, false, false);
      acc3 = __builtin_amdgcn_wmma_f32_16x16x4_f32(false, a, false, bf3,
                                                   (short)0, acc3, false, false);
    }

    // ---- bias + store; C/D layout: N = l16, M = v + 8*hlf ----
    {
      const int obase = oc * 128 + cg * 64 + l16;
      float* yp = yb + (rt * 16 + 8 * hlf) * OUTF + obase;
      store_tile(yp +  0, bias[e * OUTF + obase +  0], acc0);
      store_tile(yp + 16, bias[e * OUTF + obase + 16], acc1);
      store_tile(yp + 32, bias[e * OUTF + obase + 32], acc2);
      store_tile(yp + 48, bias[e * OUTF + obase + 48], acc3);
    }

    if (oc == 0) {
      if (wave == 0) __builtin_amdgcn_s_wait_tensorcnt(0);  // chunk 1 landed
      __syncthreads();
    }
  }
}

extern "C" void kernel_launch(void* const* d_in, const int* in_sizes, int n_in,
                              void* d_out, int out_size, void* d_ws, size_t ws_size,
                              hipStream_t stream) {
  const float* x    = (const float*)d_in[0];
  const int*   idx  = (const int*)d_in[1];
  const float* W    = (const float*)d_in[2];
  const float* bias = (const float*)d_in[3];
  float*       y    = (float*)d_out;

  static bool attr_set = false;
  if (!attr_set) {
    (void)hipFuncSetAttribute((const void*)moe_fc_wmma_tdm,
                              hipFuncAttributeMaxDynamicSharedMemorySize,
                              (int)LDS_BYTES);
    attr_set = true;
  }

  moe_fc_wmma_tdm<<<BATCH, 256, LDS_BYTES, stream>>>(x, idx, W, bias, y);
}